// geomGAT_85504208929083
// MI455X (gfx1250) — compile-verified
//
#include <hip/hip_runtime.h>
#include <hip/hip_bf16.h>
#include <math.h>

// ---------------- problem constants (match reference) ----------------
#define NNODES 50000          // 50000 = 3125 * 16  -> no row-tile bounds checks
#define NEDGES 1600000
#define NGRAPH 128
#define HIDD   67             // mlp input (64 + 3)
#define MLPH   33             // 67 // 2
#define OUTD   16

static_assert(NNODES % 16 == 0, "row tiles must be exact");

typedef __attribute__((ext_vector_type(2))) float v2f;
typedef __attribute__((ext_vector_type(8))) float v8f;

__device__ __forceinline__ float leaky02(float x) { return x > 0.0f ? x : 0.2f * x; }

// float atomic max via signed-max / unsigned-min bit trick (init must be -inf)
__device__ __forceinline__ void atomic_max_f32(float* addr, float val) {
  if (val >= 0.0f)
    atomicMax((int*)addr, __float_as_int(val));
  else
    atomicMin((unsigned int*)addr, (unsigned int)__float_as_int(val));
}

// ---------------- utility fill ----------------
__global__ void fill_f32(float* __restrict__ p, float v, long n) {
  long i = (long)blockIdx.x * blockDim.x + threadIdx.x;
  if (i < n) p[i] = v;
}

// ---------------- WMMA fp32 GEMM: H[n,F] = A[n,K] @ W[K,F] ----------------
// Compile-time K/F. One wave32 per 16x16 output tile, K stepped by 4 through
// v_wmma_f32_16x16x4_f32 (exact fp32). B is staged zero-padded in LDS (<=16KB)
// so the unrolled hot loop is: global_load_b64 (A pair) + ds_load (B pair) + wmma.
// Layouts per CDNA5 ISA:
//   A 16x4 : lane = M (mod 16), regs {K,K+1}, upper half-wave offset +2 in K.
//   B 4x16 : lane = N (mod 16), regs {K,K+1}, upper half-wave offset +2 in K.
//   C/D    : lane = N (mod 16), VGPR r -> rows r (lanes 0-15) / r+8 (lanes 16-31).
template <int K, int F>
__global__ void wmma_gemm_t(const float* __restrict__ A, const float* __restrict__ W,
                            float* __restrict__ Hout) {
  constexpr int KP = (K + 3) & ~3;       // K padded to multiple of 4
  constexpr int FP = (F + 15) & ~15;     // F padded to multiple of 16
  constexpr int colTiles = FP / 16;
  constexpr int tiles = colTiles * (NNODES / 16);

  __shared__ float Wlds[KP * FP];
  for (int i = threadIdx.x; i < KP * FP; i += blockDim.x) {
    int kk = i / FP, cc = i % FP;
    Wlds[i] = (kk < K && cc < F) ? W[kk * F + cc] : 0.0f;
  }
  __syncthreads();

  int wave = (blockIdx.x * blockDim.x + threadIdx.x) >> 5;
  if (wave >= tiles) return;             // wave-uniform: EXEC stays full for WMMA
  int rowTile = wave / colTiles, colTile = wave % colTiles;
  int row0 = rowTile << 4, col0 = colTile << 4;
  int lane = threadIdx.x & 31;
  int half = lane >> 4;
  int lm = lane & 15;
  int r = row0 + lm;                     // A row owned by this lane (always < NNODES)
  int c = col0 + lm;                     // B/D column owned by this lane

  const float* __restrict__ Arow = A + (long)r * K;
  v8f acc = {};
#pragma unroll
  for (int k = 0; k < KP; k += 4) {
    int ka = k + 2 * half;
    v2f a, b;
    if constexpr (K % 4 == 0) {
      a = *(const v2f*)(Arow + ka);      // contiguous pair -> global_load_b64
    } else {
      // K==3: ka in {0,2} always in-bounds; ka+1 may be 3 -> clamped load + select
      a.x = Arow[ka];
      float ay = Arow[ka + 1 < K ? ka + 1 : 0];
      a.y = (ka + 1 < K) ? ay : 0.0f;
    }
    b.x = Wlds[ka * FP + c];             // zero-padded: unconditional ds_load
    b.y = Wlds[(ka + 1) * FP + c];
    acc = __builtin_amdgcn_wmma_f32_16x16x4_f32(false, a, false, b,
                                                (short)0, acc, false, false);
  }

  if (c < F) {                           // only tail-column lanes masked (F==3 case)
#pragma unroll
    for (int rr = 0; rr < 8; ++rr) {
      int row = row0 + rr + 8 * half;
      Hout[(long)row * F + c] = acc[rr];
    }
  }
}

// ---------------- per-node attention logits: al = (h * a).sum(-1) ----------------
__global__ void node_logits(const float* __restrict__ hmat, const float* __restrict__ a_s,
                            const float* __restrict__ a_d, float* __restrict__ als,
                            float* __restrict__ ald, int n, int Hh, int Cc) {
  int t = blockIdx.x * blockDim.x + threadIdx.x;
  if (t >= n * Hh) return;
  int h = t % Hh;
  int node = t / Hh;
  const float* hp = hmat + (long)node * Hh * Cc + (long)h * Cc;
  float ss = 0.0f, dd = 0.0f;
  for (int cc = 0; cc < Cc; ++cc) {
    float hv = hp[cc];
    ss += hv * a_s[h * Cc + cc];
    dd += hv * a_d[h * Cc + cc];
  }
  als[t] = ss;
  ald[t] = dd;
}

// edge id -> (src, dst); edges >= NEDGES are the appended self-loops
__device__ __forceinline__ void edge_sd(const int* __restrict__ ei, long e, int& s, int& d) {
  if (e < NEDGES) { s = ei[e]; d = ei[NEDGES + e]; }
  else            { s = d = (int)(e - NEDGES); }
}

// ---------------- pass 1: segment max of leaky(alpha) over dst ----------------
__global__ void edge_max(const int* __restrict__ ei, const float* __restrict__ als,
                         const float* __restrict__ ald, float* __restrict__ m, int Hh) {
  long t = (long)blockIdx.x * blockDim.x + threadIdx.x;
  long total = (long)(NEDGES + NNODES) * Hh;
  if (t >= total) return;
  int h = (int)(t % Hh);
  long e = t / Hh;
  int s, d; edge_sd(ei, e, s, d);
  float alpha = leaky02(als[(long)s * Hh + h] + ald[(long)d * Hh + h]);
  atomic_max_f32(&m[(long)d * Hh + h], alpha);
}

// ---------------- pass 2: den = segment_sum(exp(alpha - m[dst])) ----------------
__global__ void edge_expsum(const int* __restrict__ ei, const float* __restrict__ als,
                            const float* __restrict__ ald, const float* __restrict__ m,
                            float* __restrict__ den, int Hh) {
  long t = (long)blockIdx.x * blockDim.x + threadIdx.x;
  long total = (long)(NEDGES + NNODES) * Hh;
  if (t >= total) return;
  int h = (int)(t % Hh);
  long e = t / Hh;
  int s, d; edge_sd(ei, e, s, d);
  long di = (long)d * Hh + h;
  float alpha = leaky02(als[(long)s * Hh + h] + ald[di]);
  atomicAdd(&den[di], __expf(alpha - m[di]));
}

// ---------------- pass 3: out[dst] += h[src] * coef (8 channels / thread) ----------------
__global__ void edge_aggr(const int* __restrict__ ei, const float* __restrict__ als,
                          const float* __restrict__ ald, const float* __restrict__ m,
                          const float* __restrict__ den, const float* __restrict__ hmat,
                          float* __restrict__ out, int Hh, int Cc) {
  int F = Hh * Cc;
  int chunks = (F + 7) / 8;                 // chunks never straddle a head here
  long t = (long)blockIdx.x * blockDim.x + threadIdx.x;
  long total = (long)(NEDGES + NNODES) * chunks;
  if (t >= total) return;
  int ch = (int)(t % chunks);
  long e = t / chunks;
  int s, d; edge_sd(ei, e, s, d);
  int f0 = ch * 8;
  int h = f0 / Cc;                          // constant within this thread's chunk
  long di = (long)d * Hh + h;
  float alpha = leaky02(als[(long)s * Hh + h] + ald[di]);
  float coef = __expf(alpha - m[di]) / (den[di] + 1e-16f);
  const float* hs = hmat + (long)s * F;
  __builtin_prefetch(hs + f0, 0, 3);        // global_prefetch_b8 (L2-resident row)
  int fend = f0 + 8 < F ? f0 + 8 : F;
#pragma unroll 8
  for (int f = f0; f < fend; ++f)
    atomicAdd(&out[(long)d * F + f], hs[f] * coef);
}

// ---------------- bias (+ optional relu) ----------------
__global__ void bias_relu(float* __restrict__ out, const float* __restrict__ b,
                          int n, int F, int relu) {
  long t = (long)blockIdx.x * blockDim.x + threadIdx.x;
  if (t >= (long)n * F) return;
  int f = (int)(t % F);
  float v = out[t] + b[f];
  out[t] = relu ? fmaxf(v, 0.0f) : v;
}

// ---------------- graph mean-pool accumulation ----------------
__global__ void pool_accum(const float* __restrict__ x, const float* __restrict__ p,
                           const int* __restrict__ batch, float* __restrict__ xg,
                           float* __restrict__ pg, float* __restrict__ cnt) {
  long t = (long)blockIdx.x * blockDim.x + threadIdx.x;
  if (t >= (long)NNODES * 68) return;
  int f = (int)(t % 68);
  int n = (int)(t / 68);
  int g = batch[n];
  if (f < 64)      atomicAdd(&xg[(long)g * 64 + f], x[(long)n * 64 + f]);
  else if (f < 67) atomicAdd(&pg[(long)g * 3 + (f - 64)], p[(long)n * 3 + (f - 64)]);
  else             atomicAdd(&cnt[g], 1.0f);
}

// ---------------- final MLP: [G,67] -> relu 33 -> 16 ----------------
__global__ void pool_mlp(const float* __restrict__ xg, const float* __restrict__ pg,
                         const float* __restrict__ cnt, const float* __restrict__ W1,
                         const float* __restrict__ b1, const float* __restrict__ W2,
                         const float* __restrict__ b2, float* __restrict__ out) {
  int g = blockIdx.x * blockDim.x + threadIdx.x;
  if (g >= NGRAPH) return;
  float inv = 1.0f / fmaxf(cnt[g], 1.0f);
  float hin[HIDD];
  for (int i = 0; i < 64; ++i) hin[i] = xg[(long)g * 64 + i] * inv;
  for (int i = 0; i < 3; ++i)  hin[64 + i] = pg[(long)g * 3 + i] * inv;
  float hid[MLPH];
  for (int j = 0; j < MLPH; ++j) {
    float acc = b1[j];
    for (int i = 0; i < HIDD; ++i) acc += hin[i] * W1[i * MLPH + j];
    hid[j] = fmaxf(acc, 0.0f);
  }
  for (int o = 0; o < OUTD; ++o) {
    float acc = b2[o];
    for (int j = 0; j < MLPH; ++j) acc += hid[j] * W2[j * OUTD + o];
    out[(long)g * OUTD + o] = acc;
  }
}

// ---------------- host-side GAT conv launcher (compile-time dims) ----------------
template <int K, int Hh, int Cc>
static void launch_gat(hipStream_t s, const float* xin, const float* W,
                       const float* as_, const float* ad_, const float* b,
                       bool relu, float* out, const int* ei,
                       float* hbuf, float* als, float* ald, float* m, float* den) {
  constexpr int n = NNODES;
  constexpr int F = Hh * Cc;
  constexpr int FP = (F + 15) & ~15;
  constexpr int tiles = (FP / 16) * (n / 16);
  // h = xin @ W  (fp32 WMMA, 8 waves / block)
  wmma_gemm_t<K, F><<<dim3((tiles + 7) / 8), dim3(256), 0, s>>>(xin, W, hbuf);
  // node logits
  node_logits<<<dim3((n * Hh + 255) / 256), dim3(256), 0, s>>>(
      hbuf, as_, ad_, als, ald, n, Hh, Cc);
  // init segment buffers
  fill_f32<<<dim3((n * Hh + 255) / 256), dim3(256), 0, s>>>(m, -INFINITY, (long)n * Hh);
  fill_f32<<<dim3((n * Hh + 255) / 256), dim3(256), 0, s>>>(den, 0.0f, (long)n * Hh);
  fill_f32<<<dim3((unsigned)(((long)n * F + 255) / 256)), dim3(256), 0, s>>>(out, 0.0f, (long)n * F);
  // edge passes (alpha recomputed from L2-resident node logits each pass)
  long te = (long)(NEDGES + n) * Hh;
  edge_max<<<dim3((unsigned)((te + 255) / 256)), dim3(256), 0, s>>>(ei, als, ald, m, Hh);
  edge_expsum<<<dim3((unsigned)((te + 255) / 256)), dim3(256), 0, s>>>(ei, als, ald, m, den, Hh);
  long ta = (long)(NEDGES + n) * ((F + 7) / 8);
  edge_aggr<<<dim3((unsigned)((ta + 255) / 256)), dim3(256), 0, s>>>(
      ei, als, ald, m, den, hbuf, out, Hh, Cc);
  // bias + optional relu
  bias_relu<<<dim3((unsigned)(((long)n * F + 255) / 256)), dim3(256), 0, s>>>(
      out, b, n, F, relu ? 1 : 0);
}

extern "C" void kernel_launch(void* const* d_in, const int* in_sizes, int n_in,
                              void* d_out, int out_size, void* d_ws, size_t ws_size,
                              hipStream_t stream) {
  (void)in_sizes; (void)n_in; (void)out_size; (void)ws_size;
  const float* x        = (const float*)d_in[0];
  const float* pos      = (const float*)d_in[1];
  const int*   ei       = (const int*)d_in[2];
  const int*   batch    = (const int*)d_in[3];
  /* d_in[4] edge_attr ignored (edge_dim=None) */
  const float* conv1_W  = (const float*)d_in[5];
  const float* conv1_as = (const float*)d_in[6];
  const float* conv1_ad = (const float*)d_in[7];
  const float* conv1_b  = (const float*)d_in[8];
  const float* conv2_W  = (const float*)d_in[9];
  const float* conv2_as = (const float*)d_in[10];
  const float* conv2_ad = (const float*)d_in[11];
  const float* conv2_b  = (const float*)d_in[12];
  const float* pos1_W   = (const float*)d_in[13];
  const float* pos1_as  = (const float*)d_in[14];
  const float* pos1_ad  = (const float*)d_in[15];
  const float* pos1_b   = (const float*)d_in[16];
  const float* pos2_W   = (const float*)d_in[17];
  const float* pos2_as  = (const float*)d_in[18];
  const float* pos2_ad  = (const float*)d_in[19];
  const float* pos2_b   = (const float*)d_in[20];
  const float* mlp_W1   = (const float*)d_in[21];
  const float* mlp_b1   = (const float*)d_in[22];
  const float* mlp_W2   = (const float*)d_in[23];
  const float* mlp_b2   = (const float*)d_in[24];

  // workspace layout (~59 MB of fp32)
  float* ws = (float*)d_ws;
  size_t off = 0;
  float* hbuf = ws + off; off += (size_t)NNODES * 64;   // GAT transformed features h
  float* hid  = ws + off; off += (size_t)NNODES * 64;   // relu'd intermediate (pos1 / conv1)
  float* x0   = ws + off; off += (size_t)NNODES * 64;   // x ping
  float* x1   = ws + off; off += (size_t)NNODES * 64;   // x pong
  float* p0   = ws + off; off += (size_t)NNODES * 3;    // pos ping
  float* p1   = ws + off; off += (size_t)NNODES * 3;    // pos pong
  float* als  = ws + off; off += (size_t)NNODES * 8;
  float* ald  = ws + off; off += (size_t)NNODES * 8;
  float* m    = ws + off; off += (size_t)NNODES * 8;
  float* den  = ws + off; off += (size_t)NNODES * 8;
  float* xg   = ws + off; off += (size_t)NGRAPH * 64;
  float* pg   = ws + off; off += (size_t)NGRAPH * 3;
  float* cnt  = ws + off; off += (size_t)NGRAPH;

  const float* xcur = x;
  const float* pcur = pos;
  float* xping[2] = { x0, x1 };
  float* pping[2] = { p0, p1 };

  for (int layer = 0; layer < 5; ++layer) {
    // pos = relu(posGAT1(pos))  : 3 -> 64 (H=8, C=8)
    launch_gat<3, 8, 8>(stream, pcur, pos1_W, pos1_as, pos1_ad, pos1_b, true,
                        hid, ei, hbuf, als, ald, m, den);
    // pos = posGAT2(pos)        : 64 -> 3 (H=1, C=3)
    float* pnext = pping[layer & 1];
    launch_gat<64, 1, 3>(stream, hid, pos2_W, pos2_as, pos2_ad, pos2_b, false,
                         pnext, ei, hbuf, als, ald, m, den);
    // x = relu(conv1(x))        : 64 -> 64 (H=8, C=8)
    launch_gat<64, 8, 8>(stream, xcur, conv1_W, conv1_as, conv1_ad, conv1_b, true,
                         hid, ei, hbuf, als, ald, m, den);
    // x = conv2(x)              : 64 -> 64 (H=1, C=64)
    float* xnext = xping[layer & 1];
    launch_gat<64, 1, 64>(stream, hid, conv2_W, conv2_as, conv2_ad, conv2_b, false,
                          xnext, ei, hbuf, als, ald, m, den);
    xcur = xnext;
    pcur = pnext;
  }

  // global mean pool + MLP head
  fill_f32<<<dim3((NGRAPH * 64 + 255) / 256), dim3(256), 0, stream>>>(xg, 0.0f, (long)NGRAPH * 64);
  fill_f32<<<dim3((NGRAPH * 3 + 255) / 256), dim3(256), 0, stream>>>(pg, 0.0f, (long)NGRAPH * 3);
  fill_f32<<<dim3((NGRAPH + 255) / 256), dim3(256), 0, stream>>>(cnt, 0.0f, (long)NGRAPH);
  long tp = (long)NNODES * 68;
  pool_accum<<<dim3((unsigned)((tp + 255) / 256)), dim3(256), 0, stream>>>(
      xcur, pcur, batch, xg, pg, cnt);
  pool_mlp<<<dim3(1), dim3(128), 0, stream>>>(xg, pg, cnt, mlp_W1, mlp_b1, mlp_W2, mlp_b2,
                                              (float*)d_out);
}